// PairUpdate_48524540510542
// MI455X (gfx1250) — compile-verified
//
#include <hip/hip_runtime.h>

typedef __attribute__((ext_vector_type(2))) float v2f;
typedef __attribute__((ext_vector_type(4))) float v4f;
typedef __attribute__((ext_vector_type(8))) float v8f;

#define L_SEQ 768
#define D_S   384
#define D_P   128
#define N_COLS 256  // concatenated [left|right] channel dim

// ---------------------------------------------------------------------------
// Kernel 1: LR[0:768, 0:256] = s(768x384) @ [W1 | W2](384x256), bias folded
// into the left half. One wave computes one 16x16 tile via
// v_wmma_f32_16x16x4_f32, K-loop of 96 steps.
// ---------------------------------------------------------------------------
__global__ __launch_bounds__(256) void proj_wmma_kernel(
    const float* __restrict__ s, const float* __restrict__ W,
    const float* __restrict__ bias, float* __restrict__ LR) {
  const int wave = threadIdx.x >> 5;
  const int lane = threadIdx.x & 31;
  const int tile = blockIdx.x * 8 + wave;  // 0..767
  const int tM = tile >> 4;                // 0..47  (M tiles of 16)
  const int tN = tile & 15;                // 0..15  (N tiles of 16)
  const int mBase = tM * 16;
  const int nBase = tN * 16;
  const bool hi = lane >= 16;
  const int l16 = lane & 15;

  // Wcat[k][n] : n<128 -> W[k][n] (W1), n>=128 -> W[384+k][n-128] (W2).
  // Fold into a single base pointer per wave (nBase never straddles 128).
  const float* Wbase = (tN < 8) ? W : (W + (size_t)D_S * D_P - D_P);
  const int n = nBase + l16;

  // A fragment: lanes 0-15 hold (K=k,k+1) of row mBase+l16; lanes 16-31 hold
  // (K=k+2,k+3) -> contiguous float2 with an element offset of 2 for hi lanes.
  const float* arow = s + (size_t)(mBase + l16) * D_S + (hi ? 2 : 0);
  // B fragment: lanes 0-15 hold (K=k,k+1) of column n; lanes 16-31 (K=k+2,k+3).
  const float* bcol = Wbase + n + (hi ? (size_t)2 * D_P : 0);

  v8f acc = {};
#pragma unroll 4
  for (int k = 0; k < D_S; k += 4) {
    v2f a = *(const v2f*)(arow + k);
    v2f b;
    b.x = bcol[(size_t)k * D_P];
    b.y = bcol[(size_t)(k + 1) * D_P];
    acc = __builtin_amdgcn_wmma_f32_16x16x4_f32(
        /*neg_a=*/false, a, /*neg_b=*/false, b,
        /*c_mod=*/(short)0, acc, /*reuse_a=*/false, /*reuse_b=*/false);
  }

  // Fold bias into the left half only (out = left + right + bias).
  if (tN < 8) {
    const float bv = bias[n];
#pragma unroll
    for (int v = 0; v < 8; ++v) acc[v] += bv;
  }

  // C/D layout: VGPR v -> row mBase+v (lanes 0-15) / mBase+v+8 (lanes 16-31),
  // col nBase + (lane&15).
  const int rowBase = mBase + (hi ? 8 : 0);
#pragma unroll
  for (int v = 0; v < 8; ++v)
    LR[(size_t)(rowBase + v) * N_COLS + nBase + l16] = acc[v];
}

// ---------------------------------------------------------------------------
// Kernel 2: out[i][j][f] = leftb[i][f] + right[j][f]  (302 MB, store-bound).
// One wave handles (i, 4 consecutive j), each lane owns 4 channels (float4).
// Output is non-temporal (bigger than the 192 MB L2); LR stays L2-resident.
// ---------------------------------------------------------------------------
__global__ __launch_bounds__(256) void bcast_add_kernel(
    const float* __restrict__ LR, float* __restrict__ out) {
  const int lane = threadIdx.x & 31;
  const int wave = threadIdx.x >> 5;          // 0..7
  const int i = blockIdx.y;                   // 0..767
  const int jBase = blockIdx.x * 32 + wave * 4;
  const int f = lane * 4;

  const v4f lv = *(const v4f*)(LR + (size_t)i * N_COLS + f);       // left+bias
  const float* rbase = LR + D_P + f;                               // right half
  float* obase = out + ((size_t)i * L_SEQ + jBase) * D_P + f;

#pragma unroll
  for (int q = 0; q < 4; ++q) {
    const v4f rv = *(const v4f*)(rbase + (size_t)(jBase + q) * N_COLS);
    const v4f o = lv + rv;
    __builtin_nontemporal_store(o, (v4f*)(obase + (size_t)q * D_P));
  }
}

extern "C" void kernel_launch(void* const* d_in, const int* in_sizes, int n_in,
                              void* d_out, int out_size, void* d_ws, size_t ws_size,
                              hipStream_t stream) {
  const float* s    = (const float*)d_in[0];
  // d_in[1] is z (768*768*128) — unused by the reference output.
  const float* W    = (const float*)d_in[2];
  const float* bias = (const float*)d_in[3];
  float* out = (float*)d_out;
  float* LR  = (float*)d_ws;  // 768 x 256 f32 = 786,432 bytes of scratch

  // 768 tiles / 8 waves per block
  proj_wmma_kernel<<<96, 256, 0, stream>>>(s, W, bias, LR);

  dim3 grid(L_SEQ / 32, L_SEQ);  // (24, 768) blocks of 256 threads
  bcast_add_kernel<<<grid, 256, 0, stream>>>(LR, out);
}